// CLIPModel_15006615734106
// MI455X (gfx1250) — compile-verified
//
#include <hip/hip_runtime.h>
#include <hip/hip_bf16.h>

// ---------------------------------------------------------------------------
// Problem constants (from reference): N=4096, C=768, H=512, O=50, CLASSES=1
// ---------------------------------------------------------------------------
#define NN   4096
#define CC   768
#define HH   512
#define OO   50
#define OPAD 64          // O padded to a 32-multiple-friendly tile width
#define MW   (NN / 32)   // mask words per row (128)
#define CK   64          // K-chunk staged into LDS by the TDM

typedef __attribute__((ext_vector_type(2))) float    v2f;
typedef __attribute__((ext_vector_type(8))) float    v8f;
typedef __attribute__((ext_vector_type(4))) unsigned u32x4;
typedef __attribute__((ext_vector_type(8))) int      i32x8;
typedef __attribute__((ext_vector_type(4))) int      i32x4;

// fp32 WMMA: D(16x16) += A(16x4) * B(4x16). A/B = 2 VGPRs/lane, C/D = 8.
__device__ __forceinline__ v8f wmma4(v8f acc, v2f a, v2f b) {
  return __builtin_amdgcn_wmma_f32_16x16x4_f32(
      /*neg_a=*/false, a, /*neg_b=*/false, b,
      /*c_mod=*/(short)0, acc, /*reuse_a=*/false, /*reuse_b=*/false);
}

__device__ __forceinline__ unsigned ballot32(bool p) {
  return __builtin_amdgcn_ballot_w32(p);
}

// ---------------------------------------------------------------------------
// TDM: issue a 2-D tensor_load_to_lds of a [rows=CK] x [cols<=32] fp32 tile.
//   B tile start = gaddr, row stride = ldb elements. Columns >= remw are
//   zero-filled by TDM out-of-bounds semantics (implements the O padding).
// Descriptor packed per CDNA5 ISA D# groups 0/1; groups 2/3 zero => 2-D.
// This toolchain's builtin takes 6 args (g0, g1, g2, g3, g_extra, cpol).
// ---------------------------------------------------------------------------
__device__ __forceinline__ void tdm_load_panel(const float* gptr, int ldb,
                                               int remw, unsigned lds_addr) {
  unsigned long long ga = (unsigned long long)(uintptr_t)gptr;
  if (remw > 32) remw = 32;
  if (remw < 0)  remw = 0;
  u32x4 g0;
  g0.x = 1u;                                   // count=1, user descriptor
  g0.y = lds_addr;                             // lds_addr [63:32]
  g0.z = (unsigned)(ga & 0xFFFFFFFFu);         // global_addr lo
  g0.w = (unsigned)((ga >> 32) & 0x1FFFFFFu)   // global_addr hi (57-bit)
         | (2u << 30);                         // type=2 ("image")
  i32x8 g1;
  g1[0] = (int)(2u << 16);                     // data_size=4B; wg_mask=0
  g1[1] = (int)(((unsigned)remw & 0xFFFFu) << 16);  // tensor_dim0 lo16
  g1[2] = (int)(((unsigned)CK & 0xFFFFu) << 16);    // dim0 hi=0 | tensor_dim1 lo16
  g1[3] = (int)(32u << 16);                    // dim1 hi=0 | tile_dim0=32
  g1[4] = (int)CK;                             // tile_dim1=CK | tile_dim2=0
  g1[5] = ldb;                                 // tensor_dim0_stride lo32
  g1[6] = 0;                                   // stride0 hi | stride1 lo
  g1[7] = 0;
  i32x4 gz4 = {0, 0, 0, 0};
  i32x8 gz8 = {0, 0, 0, 0, 0, 0, 0, 0};
  __builtin_amdgcn_tensor_load_to_lds(g0, g1, gz4, gz4, gz8, 0);
}

// ---------------------------------------------------------------------------
// Kernel 0: zero accumulators in workspace
// ---------------------------------------------------------------------------
__global__ void zero_ws_kernel(float* u0, float* u1, int* n01, unsigned* deg) {
  int t = blockIdx.x * blockDim.x + threadIdx.x;
  if (t < CC) { u0[t] = 0.0f; u1[t] = 0.0f; }
  if (t < 2)  { n01[t] = 0; }
  if (t < NN) { deg[t] = 0u; }
}

// ---------------------------------------------------------------------------
// Kernel 1: label-masked column sums of x and label counts.
// ---------------------------------------------------------------------------
__global__ void label_colsum_kernel(const float* __restrict__ x,
                                    const int* __restrict__ labels,
                                    float* __restrict__ u0,
                                    float* __restrict__ u1,
                                    int* __restrict__ n01) {
  const int row0 = blockIdx.x * 256;   // 16 blocks x 256 rows
  const int t = threadIdx.x;
  __shared__ int slab[256];
  __shared__ int cnt[2];
  slab[t] = labels[row0 + t];
  if (t < 2) cnt[t] = 0;
  __syncthreads();
  atomicAdd(&cnt[slab[t] != 0 ? 1 : 0], 1);
  for (int c = t; c < CC; c += 256) {
    float s0 = 0.0f, s1 = 0.0f;
    for (int rr = 0; rr < 256; ++rr) {
      float v = x[(size_t)(row0 + rr) * CC + c];
      if (slab[rr] == 0) s0 += v; else s1 += v;
    }
    atomicAdd(&u0[c], s0);
    atomicAdd(&u1[c], s1);
  }
  __syncthreads();
  if (t < 2) atomicAdd(&n01[t], cnt[t]);
}

// ---------------------------------------------------------------------------
// Kernel 2: threshold = (u0 . u1) / (n0*n1)   [== (2*u0.u1)/(2*n0*n1)]
// ---------------------------------------------------------------------------
__global__ void threshold_kernel(const float* __restrict__ u0,
                                 const float* __restrict__ u1,
                                 const int* __restrict__ n01,
                                 float* __restrict__ thr) {
  __shared__ float red[256];
  const int t = threadIdx.x;
  float acc = 0.0f;
  for (int c = t; c < CC; c += 256) acc += u0[c] * u1[c];
  red[t] = acc;
  __syncthreads();
  for (int s = 128; s > 0; s >>= 1) {
    if (t < s) red[t] += red[t + s];
    __syncthreads();
  }
  if (t == 0) {
    long long cnt = (long long)n01[0] * (long long)n01[1];
    thr[0] = (cnt > 0) ? (red[0] / (float)cnt) : 0.0f;
  }
}

// ---------------------------------------------------------------------------
// Kernel 3: sim = x x^T tiles via WMMA (32x32 per wave, 4 accumulators);
// builds bit-packed edge mask + degrees + loss_mask. sim never hits HBM.
// Grid: (NN/32, NN/32), block = 32.
// ---------------------------------------------------------------------------
__global__ void sim_mask_kernel(const float* __restrict__ x,
                                const int* __restrict__ labels,
                                const float* __restrict__ thr_p,
                                unsigned* __restrict__ mask,
                                unsigned* __restrict__ deg,
                                float* __restrict__ lossm) {
  const int lane = threadIdx.x & 31;
  const int jt = blockIdx.x;            // 32-col tile
  const int it = blockIdx.y;            // 32-row tile
  const int I0 = it * 32, J0 = jt * 32;
  const int r  = lane & 15;
  const int kh = (lane >> 4) << 1;      // lanes 0-15: k+0/1, lanes 16-31: k+2/3

  const float* arow0 = x + (size_t)(I0 + r) * CC + kh;
  const float* arow1 = x + (size_t)(I0 + 16 + r) * CC + kh;
  const float* brow0 = x + (size_t)(J0 + r) * CC + kh;
  const float* brow1 = x + (size_t)(J0 + 16 + r) * CC + kh;

  v8f acc[2][2] = {{{}, {}}, {{}, {}}};
  for (int kb = 0; kb < CC; kb += 4) {
    v2f a0 = *(const v2f*)(arow0 + kb);
    v2f a1 = *(const v2f*)(arow1 + kb);
    v2f b0 = *(const v2f*)(brow0 + kb);
    v2f b1 = *(const v2f*)(brow1 + kb);
    acc[0][0] = wmma4(acc[0][0], a0, b0);
    acc[0][1] = wmma4(acc[0][1], a0, b1);
    acc[1][0] = wmma4(acc[1][0], a1, b0);
    acc[1][1] = wmma4(acc[1][1], a1, b1);
  }

  const float thr = *thr_p;
  const int jc0 = J0 + r;          // this lane's column (left 16)
  const int jc1 = jc0 + 16;        // this lane's column (right 16)
  const int lj0 = labels[jc0];
  const int lj1 = labels[jc1];

#pragma unroll
  for (int g = 0; g < 2; ++g) {
    const int rowbase = I0 + g * 16;
    const int ibase = rowbase + ((lane >> 4) << 3);  // hi half holds M=v+8
#pragma unroll
    for (int v = 0; v < 8; ++v) {
      const int i = ibase + v;
      const int li = labels[i];
      const bool p0 = (li == lj0) && (i != jc0) && (acc[g][0][v] <= thr);
      const bool p1 = (li == lj1) && (i != jc1) && (acc[g][1][v] <= thr);
      const unsigned b0 = ballot32(p0);  // [0:16)=row rowbase+v, [16:32)=row rowbase+v+8
      const unsigned b1 = ballot32(p1);
      lossm[(size_t)i * NN + jc0] = (p0 && (i < jc0)) ? 1.0f : 0.0f;
      lossm[(size_t)i * NN + jc1] = (p1 && (i < jc1)) ? 1.0f : 0.0f;
      if (lane == 0) {
        const int row_lo = rowbase + v, row_hi = rowbase + v + 8;
        const unsigned w_lo = (b0 & 0xFFFFu) | ((b1 & 0xFFFFu) << 16);
        const unsigned w_hi = (b0 >> 16)     | (b1 & 0xFFFF0000u);
        mask[(size_t)row_lo * MW + jt] = w_lo;
        mask[(size_t)row_hi * MW + jt] = w_hi;
        atomicAdd(&deg[row_lo], (unsigned)__popc(w_lo));
        atomicAdd(&deg[row_hi], (unsigned)__popc(w_hi));
      }
    }
  }
}

// ---------------------------------------------------------------------------
// Kernel 4: dinv[i] = rsqrt(1 + deg[i])   (self-loop included)
// ---------------------------------------------------------------------------
__global__ void dinv_kernel(const unsigned* __restrict__ deg,
                            float* __restrict__ dinv) {
  int i = blockIdx.x * 256 + threadIdx.x;
  if (i < NN) dinv[i] = rsqrtf(1.0f + (float)deg[i]);
}

// ---------------------------------------------------------------------------
// Kernel 5: fp32 WMMA GEMM, C = act(A[M,K]*B[K,N] + bias), with the B panel
// TDM-staged into LDS (double buffered). Block = 4 waves = 64 rows x 32 cols.
// Columns >= validN of B come back zero from TDM OOB handling.
// ---------------------------------------------------------------------------
__global__ void gemm_tdm_kernel(const float* __restrict__ A, int lda,
                                const float* __restrict__ B, int ldb,
                                const float* __restrict__ bias,
                                float* __restrict__ Cout, int ldc,
                                int K, int validN, int relu) {
  __shared__ float bpanel[2][CK][32];
  const int tid  = threadIdx.x;
  const int lane = tid & 31;
  const int wave = tid >> 5;
  const int J0 = blockIdx.x * 32;
  const int I0 = blockIdx.y * 64 + wave * 16;
  const int r  = lane & 15;
  const int kh = (lane >> 4) << 1;
  const int j0l = r, j1l = r + 16;
  const int nchunk = K / CK;
  const int remw = validN - J0;

  const float* arow = A + (size_t)(I0 + r) * lda + kh;
  v8f c0 = {}, c1 = {};

  if (wave == 0) {
    tdm_load_panel(B + (size_t)0 * ldb + J0, ldb, remw,
                   (unsigned)(uintptr_t)&bpanel[0][0][0]);
  }
  for (int c = 0; c < nchunk; ++c) {
    if (wave == 0) __builtin_amdgcn_s_wait_tensorcnt(0);  // chunk c landed
    __syncthreads();  // data visible; prior reads of the other buffer done
    if (wave == 0 && c + 1 < nchunk) {
      tdm_load_panel(B + (size_t)(c + 1) * CK * ldb + J0, ldb, remw,
                     (unsigned)(uintptr_t)&bpanel[(c + 1) & 1][0][0]);
    }
    const float(*bp)[32] = bpanel[c & 1];
    const int kbase = c * CK;
#pragma unroll 4
    for (int kk = 0; kk < CK; kk += 4) {
      const int kp = kk + kh;
      v2f a = *(const v2f*)(arow + kbase + kk);
      v2f b0, b1;
      b0.x = bp[kp][j0l];
      b0.y = bp[kp + 1][j0l];
      b1.x = bp[kp][j1l];
      b1.y = bp[kp + 1][j1l];
      c0 = wmma4(c0, a, b0);
      c1 = wmma4(c1, a, b1);
    }
  }
  const int j0 = J0 + r, j1 = j0 + 16;
  const float bb0 = (bias && (j0 < validN)) ? bias[j0] : 0.0f;
  const float bb1 = (bias && (j1 < validN)) ? bias[j1] : 0.0f;
  const int ibase = I0 + ((lane >> 4) << 3);
#pragma unroll
  for (int v = 0; v < 8; ++v) {
    float d0 = c0[v] + bb0;
    float d1 = c1[v] + bb1;
    if (relu) { d0 = fmaxf(d0, 0.0f); d1 = fmaxf(d1, 0.0f); }
    Cout[(size_t)(ibase + v) * ldc + j0] = d0;
    Cout[(size_t)(ibase + v) * ldc + j1] = d1;
  }
}

// ---------------------------------------------------------------------------
// Kernel 6: C = act(norm_adj * B + bias), norm_adj regenerated on the fly:
//   A[i][k] = dinv[i]*dinv[k] * (maskbit(i,k) | (i==k)),   K = NN.
// B panel TDM-staged into LDS exactly like gemm_tdm_kernel.
// ---------------------------------------------------------------------------
__global__ void adj_gemm_tdm_kernel(const unsigned* __restrict__ mask,
                                    const float* __restrict__ dinv,
                                    const float* __restrict__ B, int ldb,
                                    const float* __restrict__ bias,
                                    float* __restrict__ Cout, int ldc,
                                    int bN /*valid B cols for TDM*/,
                                    int biasN, int relu) {
  __shared__ float bpanel[2][CK][32];
  const int tid  = threadIdx.x;
  const int lane = tid & 31;
  const int wave = tid >> 5;
  const int J0 = blockIdx.x * 32;
  const int I0 = blockIdx.y * 64 + wave * 16;
  const int r  = lane & 15;
  const int kh = (lane >> 4) << 1;
  const int j0l = r, j1l = r + 16;
  const int i  = I0 + r;
  const float di = dinv[i];
  const unsigned* mrow = mask + (size_t)i * MW;
  const int nchunk = NN / CK;
  const int remw = bN - J0;

  v8f c0 = {}, c1 = {};
  if (wave == 0) {
    tdm_load_panel(B + J0, ldb, remw, (unsigned)(uintptr_t)&bpanel[0][0][0]);
  }
  for (int c = 0; c < nchunk; ++c) {
    if (wave == 0) __builtin_amdgcn_s_wait_tensorcnt(0);
    __syncthreads();
    if (wave == 0 && c + 1 < nchunk) {
      tdm_load_panel(B + (size_t)(c + 1) * CK * ldb + J0, ldb, remw,
                     (unsigned)(uintptr_t)&bpanel[(c + 1) & 1][0][0]);
    }
    const float(*bp)[32] = bpanel[c & 1];
    const int kbase = c * CK;
#pragma unroll 4
    for (int kk = 0; kk < CK; kk += 4) {
      const int kp = kbase + kk + kh;
      const unsigned w  = mrow[kp >> 5];
      const unsigned sh = (unsigned)(kp & 31);
      const float m0 = (float)(((w >> sh) & 1u)       | (unsigned)(i == kp));
      const float m1 = (float)(((w >> (sh + 1)) & 1u) | (unsigned)(i == kp + 1));
      v2f a;
      a.x = di * dinv[kp] * m0;
      a.y = di * dinv[kp + 1] * m1;
      const int kl = kk + kh;
      v2f b0, b1;
      b0.x = bp[kl][j0l];
      b0.y = bp[kl + 1][j0l];
      b1.x = bp[kl][j1l];
      b1.y = bp[kl + 1][j1l];
      c0 = wmma4(c0, a, b0);
      c1 = wmma4(c1, a, b1);
    }
  }
  const int j0 = J0 + r, j1 = j0 + 16;
  const float bb0 = (bias && (j0 < biasN)) ? bias[j0] : 0.0f;
  const float bb1 = (bias && (j1 < biasN)) ? bias[j1] : 0.0f;
  const int ibase = I0 + ((lane >> 4) << 3);
#pragma unroll
  for (int v = 0; v < 8; ++v) {
    float d0 = c0[v] + bb0;
    float d1 = c1[v] + bb1;
    if (relu) { d0 = fmaxf(d0, 0.0f); d1 = fmaxf(d1, 0.0f); }
    Cout[(size_t)(ibase + v) * ldc + j0] = d0;
    Cout[(size_t)(ibase + v) * ldc + j1] = d1;
  }
}

// ---------------------------------------------------------------------------
// Kernel 7: fused final stage. Per row i:
//   f_g[i] = concat(x[i], g[i]);  out[i] = f_g[i] . fcW + fcb;  x_out[i] = x[i]
// ---------------------------------------------------------------------------
__global__ void fc_concat_kernel(const float* __restrict__ x,
                                 const float* __restrict__ g,   // stride OPAD
                                 const float* __restrict__ fcW,
                                 const float* __restrict__ fcb,
                                 float* __restrict__ out,
                                 float* __restrict__ f_g,
                                 float* __restrict__ x_out) {
  const int i = blockIdx.x;
  const int t = threadIdx.x;
  __shared__ float red[256];
  float acc = 0.0f;
  for (int c = t; c < CC; c += 256) {
    float v = x[(size_t)i * CC + c];
    f_g[(size_t)i * (CC + OO) + c] = v;
    x_out[(size_t)i * CC + c] = v;
    acc += v * fcW[c];
  }
  for (int o = t; o < OO; o += 256) {
    float v = g[(size_t)i * OPAD + o];
    f_g[(size_t)i * (CC + OO) + CC + o] = v;
    acc += v * fcW[CC + o];
  }
  red[t] = acc;
  __syncthreads();
  for (int s = 128; s > 0; s >>= 1) {
    if (t < s) red[t] += red[t + s];
    __syncthreads();
  }
  if (t == 0) out[i] = red[0] + fcb[0];
}

// ---------------------------------------------------------------------------
// Host launch
// ---------------------------------------------------------------------------
extern "C" void kernel_launch(void* const* d_in, const int* in_sizes, int n_in,
                              void* d_out, int out_size, void* d_ws, size_t ws_size,
                              hipStream_t stream) {
  const float* x      = (const float*)d_in[0];
  const int*   labels = (const int*)d_in[1];
  const float* W1     = (const float*)d_in[2];
  const float* b1     = (const float*)d_in[3];
  const float* W2     = (const float*)d_in[4];
  const float* b2     = (const float*)d_in[5];
  const float* fcW    = (const float*)d_in[6];
  const float* fcb    = (const float*)d_in[7];

  // Workspace carve-out (256B aligned slices)
  char* ws = (char*)d_ws;
  size_t o = 0;
  auto carve = [&](size_t bytes) -> void* {
    o = (o + 255) & ~(size_t)255;
    void* p = ws + o;
    o += bytes;
    return p;
  };
  float*    u0   = (float*)carve(CC * sizeof(float));
  float*    u1   = (float*)carve(CC * sizeof(float));
  int*      n01  = (int*)carve(2 * sizeof(int));
  float*    thr  = (float*)carve(sizeof(float));
  unsigned* deg  = (unsigned*)carve(NN * sizeof(unsigned));
  float*    dinv = (float*)carve(NN * sizeof(float));
  unsigned* mask = (unsigned*)carve((size_t)NN * MW * sizeof(unsigned)); // 2 MB
  float*    P    = (float*)carve((size_t)NN * HH * sizeof(float));       // 8 MB
  float*    hbuf = (float*)carve((size_t)NN * HH * sizeof(float));       // 8 MB
  float*    qbuf = (float*)carve((size_t)NN * OPAD * sizeof(float));     // 1 MB
  float*    gbuf = (float*)carve((size_t)NN * OPAD * sizeof(float));     // 1 MB

  // Output layout: [out | f_g | loss_mask | x] flattened (reference order)
  float* out_p   = (float*)d_out;
  float* f_g_p   = out_p + NN;
  float* loss_p  = f_g_p + (size_t)NN * (CC + OO);
  float* x_out_p = loss_p + (size_t)NN * NN;

  // 0) zero accumulators
  zero_ws_kernel<<<dim3(16), dim3(256), 0, stream>>>(u0, u1, n01, deg);
  // 1) label-masked column sums + counts
  label_colsum_kernel<<<dim3(16), dim3(256), 0, stream>>>(x, labels, u0, u1, n01);
  // 2) threshold scalar
  threshold_kernel<<<dim3(1), dim3(256), 0, stream>>>(u0, u1, n01, thr);
  // 3) sim (WMMA 32x32/wave) -> bitmask + degrees + loss_mask output
  sim_mask_kernel<<<dim3(NN / 32, NN / 32), dim3(32), 0, stream>>>(
      x, labels, thr, mask, deg, loss_p);
  // 4) dinv = rsqrt(1 + deg)
  dinv_kernel<<<dim3(NN / 256), dim3(256), 0, stream>>>(deg, dinv);
  // 5) P = x @ W1                            [4096 x 512]
  gemm_tdm_kernel<<<dim3(HH / 32, NN / 64), dim3(128), 0, stream>>>(
      x, CC, W1, HH, nullptr, P, HH, CC, HH, 0);
  // 6) h = relu(norm_adj @ P + b1)           [4096 x 512]
  adj_gemm_tdm_kernel<<<dim3(HH / 32, NN / 64), dim3(128), 0, stream>>>(
      mask, dinv, P, HH, b1, hbuf, HH, HH, HH, 1);
  // 7) q = h @ W2 (cols padded 50->64 via TDM OOB zero-fill)  [4096 x 64]
  gemm_tdm_kernel<<<dim3(OPAD / 32, NN / 64), dim3(128), 0, stream>>>(
      hbuf, HH, W2, OO, nullptr, qbuf, OPAD, HH, OO, 0);
  // 8) g = norm_adj @ q + b2                 [4096 x 64]
  adj_gemm_tdm_kernel<<<dim3(OPAD / 32, NN / 64), dim3(128), 0, stream>>>(
      mask, dinv, qbuf, OPAD, b2, gbuf, OPAD, OPAD, OO, 0);
  // 9) out / f_g / x copies
  fc_concat_kernel<<<dim3(NN), dim3(256), 0, stream>>>(
      x, gbuf, fcW, fcb, out_p, f_g_p, x_out_p);
}